// FeaturePlanes_65506841199149
// MI455X (gfx1250) — compile-verified
//
#include <hip/hip_runtime.h>
#include <math.h>
#include <stdint.h>

// ---------------------------------------------------------------------------
// FeaturePlanes sampler for MI455X (gfx1250, wave32).
//
// Phase 1 (repack): fm [P][C=16][S][S] -> buf [P][S][S][C=16] so each bilinear
//   corner fetch of 16 channels is one contiguous 64B segment. The strided
//   16-row tile load is done by the Tensor Data Mover (tensor_load_to_lds,
//   tensor_dim0_stride = S*S, pad feature -> LDS row stride 260 DW to dodge
//   bank conflicts), synchronized with s_wait_tensorcnt. Manual coalesced
//   fallback if the builtin is unavailable.
// Phase 2 (sample): 4 lanes/point, 4 channels/lane -> global_load_b128 per
//   corner, global_prefetch_b8 to warm L2, non-temporal b128 output stores so
//   the 384MB write-once output doesn't evict the feature planes from L2.
// ---------------------------------------------------------------------------

typedef float v4f __attribute__((ext_vector_type(4)));

__device__ __forceinline__ float lerpf(float a, float b, float w) {
    return fmaf(w, b - a, a);  // a + w*(b-a)
}

// align_corners=True, border padding: px = clip((v+1)*(S-1)/2, 0, S-1);
// fold x1=min(x0+1,S-1) into x0=min(x0,S-2) (weight absorbs the edge case).
__device__ __forceinline__ void coord4(v4f xv, int S, int* i0, float* w) {
    const float scale = 0.5f * (float)(S - 1);
    const float hi    = (float)(S - 1);
    float v[4] = {xv.x, xv.y, xv.z, xv.w};
#pragma unroll
    for (int d = 0; d < 4; ++d) {
        float p = (v[d] + 1.0f) * scale;
        p = fminf(fmaxf(p, 0.0f), hi);
        int q = (int)p;                 // floor (p >= 0)
        if (q > S - 2) q = S - 2;       // keep [q, q+1] in range; w becomes 1
        i0[d] = q;
        w[d]  = p - (float)q;
    }
}

// ---------------------------------------------------------------------------
// Repack tile kernel: one block = one (plane, 256-position chunk).
// LDS tile: 16 channel-rows x 256 positions, row stride 260 DWORDs.
// ---------------------------------------------------------------------------
#define TDM_CHUNK 256
#define TDM_LDSW  260

__global__ __launch_bounds__(256) void fp_repack_tile(
    const float* __restrict__ fm, float* __restrict__ buf, long long SS) {
    __shared__ float tile[16 * TDM_LDSW];

    const int       p    = blockIdx.y;
    const long long pos0 = (long long)blockIdx.x * TDM_CHUNK;
    const int       t    = threadIdx.x;

#if defined(__HIP_DEVICE_COMPILE__) && __has_builtin(__builtin_amdgcn_tensor_load_to_lds)
    // ---- TDM path: DMA the strided 2D tile [16 ch x 256 pos] into LDS ----
    if (t < 32) {  // wave 0 issues the DMA (wave-level op; EXEC ignored)
        typedef unsigned int tdm_v4u __attribute__((ext_vector_type(4)));
        typedef int          tdm_v8i __attribute__((ext_vector_type(8)));
        typedef int          tdm_v4i __attribute__((ext_vector_type(4)));

        const unsigned long long gaddr =
            (unsigned long long)(uintptr_t)(fm + (long long)p * 16 * SS + pos0);
        const unsigned lds_off = (unsigned)(uintptr_t)(void*)tile;
        const unsigned dim0    = (unsigned)(SS - pos0);  // tile-relative length; OOB reads -> 0
        const unsigned long long stride0 = (unsigned long long)SS;

        // D# group 0: count=1 | lds_addr | 57-bit global_addr | type=2
        tdm_v4u g0 = {
            1u,
            lds_off,
            (unsigned)(gaddr & 0xFFFFFFFFu),
            (unsigned)((gaddr >> 32) & 0x1FFFFFFu) | (2u << 30)
        };
        // D# group 1:
        //  w0: mask=0 | data_size=4B(code 2)<<16 | pad_enable<<20 |
        //      pad_interval=256DW(code 7)<<22 | pad_amount=4DW(code 3)<<25
        //  dims: tensor_dim0=dim0, tensor_dim1=16, tile_dim0=256, tile_dim1=16,
        //        tile_dim2=0, tensor_dim0_stride=SS, tensor_dim1_stride=0
        const unsigned w0 = (2u << 16) | (1u << 20) | (7u << 22) | (3u << 25);
        const unsigned w1 = (dim0 & 0xFFFFu) << 16;                  // atomic_addr=0 | dim0 lo16
        const unsigned w2 = ((dim0 >> 16) & 0xFFFFu) | (16u << 16);  // dim0 hi16 | dim1 lo16
        const unsigned w3 = ((unsigned)TDM_CHUNK) << 16;             // dim1 hi16=0 | tile_dim0
        const unsigned w4 = 16u;                                     // tile_dim1 | tile_dim2=0
        const unsigned w5 = (unsigned)(stride0 & 0xFFFFFFFFull);     // dim0_stride lo32
        const unsigned w6 = (unsigned)((stride0 >> 32) & 0xFFFFull); // hi16 | dim1_stride lo=0
        tdm_v8i g1 = {(int)w0, (int)w1, (int)w2, (int)w3,
                      (int)w4, (int)w5, (int)w6, 0};
        tdm_v4i gz = {0, 0, 0, 0};  // groups 2/3 unused (2D tensor)
#if __clang_major__ >= 23
        tdm_v8i gz8 = {0, 0, 0, 0, 0, 0, 0, 0};
        __builtin_amdgcn_tensor_load_to_lds(g0, g1, gz, gz, gz8, 0);
#else
        __builtin_amdgcn_tensor_load_to_lds(g0, g1, gz, gz, 0);
#endif
        __builtin_amdgcn_s_wait_tensorcnt(0);
    }
#else
    // ---- Manual fallback: coalesced b32 loads into the same LDS layout ----
    {
        const int cl = t >> 4;   // channel 0..15
        const int il = t & 15;   // position 0..15 (stepped by 16)
        const float* src = fm + ((long long)p * 16 + cl) * SS;
#pragma unroll
        for (int k = 0; k < 16; ++k) {
            long long pos = pos0 + il + 16 * k;
            tile[cl * TDM_LDSW + il + 16 * k] =
                (pos < SS) ? __builtin_nontemporal_load(&src[pos]) : 0.0f;
        }
    }
#endif
    __syncthreads();

    // Transposed write-out: 4 lanes/position, b128 stores, contiguous across
    // the block. LDS reads hit distinct banks (row stride 260 -> 4*c+i mod 64).
    const int cq = (t & 3) << 2;   // channel quad
    const int ib = t >> 2;         // 0..63
    float* dst = buf + (long long)p * SS * 16;
#pragma unroll
    for (int k = 0; k < 4; ++k) {
        const int i   = ib + 64 * k;       // position in chunk
        long long pos = pos0 + i;
        if (pos < SS) {
            v4f v;
            v.x = tile[(cq + 0) * TDM_LDSW + i];
            v.y = tile[(cq + 1) * TDM_LDSW + i];
            v.z = tile[(cq + 2) * TDM_LDSW + i];
            v.w = tile[(cq + 3) * TDM_LDSW + i];
            *reinterpret_cast<v4f*>(dst + pos * 16 + cq) = v;
        }
    }
}

// ---------------------------------------------------------------------------
// Sampling kernel, repacked layout. 4 lanes/point, 4 channels/lane.
// ---------------------------------------------------------------------------
__global__ __launch_bounds__(256) void fp_sample_packed(
    const float* __restrict__ x, const float* __restrict__ buf,
    float* __restrict__ out, int N, int S, long long SS) {
    const int t = threadIdx.x;
    const int n = blockIdx.x * 64 + (t >> 2);
    if (n >= N) return;
    const int cq = (t & 3) << 2;  // channel quad: 0,4,8,12

    // x is streamed once -> non-temporal, keep L2 for the feature planes.
    const v4f xv = __builtin_nontemporal_load(reinterpret_cast<const v4f*>(x) + n);
    int   i0[4];
    float w[4];
    coord4(xv, S, i0, w);

    const int PI[6] = {0, 0, 0, 1, 1, 2};  // px source dim (width)
    const int PJ[6] = {1, 2, 3, 2, 3, 3};  // py source dim (height)
    const long long rowStride = (long long)S * 16;

    const float* base[6];
#pragma unroll
    for (int p = 0; p < 6; ++p) {
        base[p] = buf + ((long long)p * SS +
                         (long long)i0[PJ[p]] * S + i0[PI[p]]) * 16 + cq;
        // gfx1250 global_prefetch_b8: warm corner lines before the load burst.
        __builtin_prefetch(base[p], 0, 3);
    }

    float* orow = out + (long long)n * 96 + cq;
#pragma unroll
    for (int p = 0; p < 6; ++p) {
        const float wx = w[PI[p]];
        const float wy = w[PJ[p]];
        const v4f v00 = *reinterpret_cast<const v4f*>(base[p]);
        const v4f v01 = *reinterpret_cast<const v4f*>(base[p] + 16);
        const v4f v10 = *reinterpret_cast<const v4f*>(base[p] + rowStride);
        const v4f v11 = *reinterpret_cast<const v4f*>(base[p] + rowStride + 16);
        v4f r;
        r.x = lerpf(lerpf(v00.x, v01.x, wx), lerpf(v10.x, v11.x, wx), wy);
        r.y = lerpf(lerpf(v00.y, v01.y, wx), lerpf(v10.y, v11.y, wx), wy);
        r.z = lerpf(lerpf(v00.z, v01.z, wx), lerpf(v10.z, v11.z, wx), wy);
        r.w = lerpf(lerpf(v00.w, v01.w, wx), lerpf(v10.w, v11.w, wx), wy);
        // Output is write-once -> non-temporal store (don't pollute L2).
        __builtin_nontemporal_store(r, reinterpret_cast<v4f*>(orow + p * 16));
    }
}

// ---------------------------------------------------------------------------
// Fallback: sample directly from channel-major fm (if d_ws too small).
// ---------------------------------------------------------------------------
__global__ __launch_bounds__(256) void fp_sample_direct(
    const float* __restrict__ x, const float* __restrict__ fm,
    float* __restrict__ out, int N, int S, long long SS) {
    const int t = threadIdx.x;
    const int n = blockIdx.x * 16 + (t >> 4);
    if (n >= N) return;
    const int c = t & 15;

    const v4f xv = __builtin_nontemporal_load(reinterpret_cast<const v4f*>(x) + n);
    int   i0[4];
    float w[4];
    coord4(xv, S, i0, w);

    const int PI[6] = {0, 0, 0, 1, 1, 2};
    const int PJ[6] = {1, 2, 3, 2, 3, 3};

    float* orow = out + (long long)n * 96 + c;
#pragma unroll
    for (int p = 0; p < 6; ++p) {
        const int   xi = i0[PI[p]];
        const int   yi = i0[PJ[p]];
        const float wx = w[PI[p]];
        const float wy = w[PJ[p]];
        const float* pl = fm + ((long long)p * 16 + c) * SS +
                          (long long)yi * S + xi;
        const float v00 = pl[0];
        const float v01 = pl[1];
        const float v10 = pl[S];
        const float v11 = pl[S + 1];
        const float r = lerpf(lerpf(v00, v01, wx), lerpf(v10, v11, wx), wy);
        __builtin_nontemporal_store(r, orow + p * 16);
    }
}

extern "C" void kernel_launch(void* const* d_in, const int* in_sizes, int n_in,
                              void* d_out, int out_size, void* d_ws, size_t ws_size,
                              hipStream_t stream) {
    (void)n_in; (void)out_size;
    const float* x   = (const float*)d_in[0];
    const float* fm  = (const float*)d_in[1];
    float*       out = (float*)d_out;

    const int       N       = in_sizes[0] / 4;           // [N,4] coords
    const long long fmElems = (long long)in_sizes[1];    // P*C*S*S = 96*S*S
    const long long SS      = fmElems / 96;              // S*S
    const int       S       = (int)(sqrt((double)SS) + 0.5);
    const int       P       = (int)(fmElems / (SS * 16));

    const size_t needBytes = (size_t)fmElems * sizeof(float);
    if (d_ws != nullptr && ws_size >= needBytes) {
        float* buf = (float*)d_ws;
        dim3 gRepack((unsigned)((SS + TDM_CHUNK - 1) / TDM_CHUNK), (unsigned)P);
        fp_repack_tile<<<gRepack, 256, 0, stream>>>(fm, buf, SS);
        const int blocks = (N + 63) / 64;
        fp_sample_packed<<<blocks, 256, 0, stream>>>(x, buf, out, N, S, SS);
    } else {
        const int blocks = (N + 15) / 16;
        fp_sample_direct<<<blocks, 256, 0, stream>>>(x, fm, out, N, S, SS);
    }
}